// PrecondWL_24764781429524
// MI455X (gfx1250) — compile-verified
//
#include <hip/hip_runtime.h>
#include <hip/hip_bf16.h>

// PrecondWL: per-node wirelength preconditioner.
// out[n] = sum over pins p of node n of  clamp(w[net(p)],min=1) / max(deg(net)-1,1)
// when deg(net) > 1, movable nodes only (n < num_movable); else 0.
//
// Memory-bound gather workload: ~90MB cold DRAM traffic, ~80MB working set
// (fits in MI455X's 192MB L2) -> ~4us roofline floor at 23.3 TB/s.
// No matmul structure -> WMMA inapplicable. Strategy: 1 thread/node,
// global_load_b128 on the aligned degree-4 fast path (wave32 -> 512B
// contiguous per wave per instruction), branchless inner body (v_cndmask
// instead of exec-mask divergence), v_rcp_f32 instead of the full precise
// division sequence (divisor is an exact small integer; ~2 ulp), and a
// clamped global_prefetch_b8 to keep the streamed pin array ahead.

__global__ void __launch_bounds__(256)
precond_wl_kernel(const float* __restrict__ net_weights,
                  const int*   __restrict__ n2p_start,
                  const int*   __restrict__ n2p,
                  const int*   __restrict__ pin2net,
                  const int*   __restrict__ net2pin,
                  const int*   __restrict__ num_movable_ptr,
                  float*       __restrict__ out,
                  int num_nodes,
                  int num_pins) {
    const int n = blockIdx.x * blockDim.x + threadIdx.x;
    if (n >= num_nodes) return;

    // Uniform device-resident scalar -> scalar load.
    const int num_movable = *num_movable_ptr;

    float sum = 0.0f;
    if (n < num_movable) {
        // Adjacent starts -> compiler merges into one global_load_b64.
        const int s0 = n2p_start[n];
        const int s1 = n2p_start[n + 1];

        // Branchless clamped prefetch of the streamed pin array ~16KB ahead
        // (gfx1250 global_prefetch_b8: no VGPR return, no counter traffic).
        int pf = s0 + 4096;
        pf = (pf < num_pins) ? pf : (num_pins - 1);
        __builtin_prefetch(n2p + pf, 0, 0);

        // Non-short-circuit predicate: single compare+and, no extra branch.
        const bool fast = ((s1 - s0) == 4) & ((s0 & 3) == 0);
        if (fast) {
            // Constant degree 4, 16B-aligned -> one global_load_b128.
            const int4 pins = *reinterpret_cast<const int4*>(n2p + s0);
            const int pinv[4] = {pins.x, pins.y, pins.z, pins.w};
#pragma unroll
            for (int k = 0; k < 4; ++k) {
                const int net = pin2net[pinv[k]];   // L2-resident gather
                const int d0  = net2pin[net];
                const int d1  = net2pin[net + 1];
                const int deg = d1 - d0;
                // Branchless: always gather weight, always rcp, select at end.
                const float w  = fmaxf(net_weights[net], 1.0f);
                const int   dm = (deg - 1 > 1) ? (deg - 1) : 1;
                const float r  = __builtin_amdgcn_rcpf((float)dm);
                sum += (deg > 1) ? (w * r) : 0.0f;
            }
        } else {
            // Generic CSR fallback (kept correct for arbitrary degree).
            for (int p = s0; p < s1; ++p) {
                const int net = pin2net[n2p[p]];
                const int deg = net2pin[net + 1] - net2pin[net];
                const float w  = fmaxf(net_weights[net], 1.0f);
                const int   dm = (deg - 1 > 1) ? (deg - 1) : 1;
                const float r  = __builtin_amdgcn_rcpf((float)dm);
                sum += (deg > 1) ? (w * r) : 0.0f;
            }
        }
    }
    // Write every node (zeros for fixed nodes) -> fully initializes d_out.
    out[n] = sum;
}

extern "C" void kernel_launch(void* const* d_in, const int* in_sizes, int n_in,
                              void* d_out, int out_size, void* d_ws, size_t ws_size,
                              hipStream_t stream) {
    // setup_inputs() order:
    //   0: net_weights         float32 [N_NETS]
    //   1: flat_node2pin_start int32   [N_NODES+1]
    //   2: flat_node2pin       int32   [N_PINS]
    //   3: pin2net_map         int32   [N_PINS]
    //   4: flat_net2pin        int32   [N_NETS+1]
    //   5: num_movable_nodes   int32   [1] (device scalar)
    const float* net_weights     = (const float*)d_in[0];
    const int*   n2p_start       = (const int*)  d_in[1];
    const int*   n2p             = (const int*)  d_in[2];
    const int*   pin2net         = (const int*)  d_in[3];
    const int*   net2pin         = (const int*)  d_in[4];
    const int*   num_movable_ptr = (const int*)  d_in[5];

    const int num_nodes = in_sizes[1] - 1;
    const int num_pins  = in_sizes[2];
    float* out = (float*)d_out;

    const int block = 256;                       // 8 wave32s per block
    const int grid  = (num_nodes + block - 1) / block;

    precond_wl_kernel<<<grid, block, 0, stream>>>(
        net_weights, n2p_start, n2p, pin2net, net2pin,
        num_movable_ptr, out, num_nodes, num_pins);
}